// SGRUCell_89696097009960
// MI455X (gfx1250) — compile-verified
//
#include <hip/hip_runtime.h>

// Problem dims (from reference)
#define T_STEPS 128
#define B       32
#define IN_DIM  256
#define H       512
#define H3      1536
#define R       128
#define MODN    131      // R + 3
#define KTOP    12
#define NWG     32       // one workgroup per batch element (state owner)
#define NTHREADS 256     // 8 wave32s
#define NWAVES_TOTAL (NWG * (NTHREADS / 32))

typedef float v2f __attribute__((ext_vector_type(2)));
typedef float v8f __attribute__((ext_vector_type(8)));
typedef unsigned u32x4 __attribute__((ext_vector_type(4)));
typedef int i32x4 __attribute__((ext_vector_type(4)));
typedef int i32x8 __attribute__((ext_vector_type(8)));

// LDS staging with TDM padding: 2 DWORDs inserted every 256 DWORDs streamed.
// element at stream position p lives at float offset p + 2*(p>>8); p is even at
// every fragment load -> 8B alignment preserved; row stride becomes 516/129
// floats -> A-fragment lanes hit distinct LDS banks.
#define PADIDX(p) ((p) + 2 * ((p) >> 8))
#define HB_ELEMS  (B * H + 2 * ((B * H) >> 8))     // 16512 floats
#define AB_ELEMS  (B * R + 2 * ((B * R) >> 8))     // 4128 floats

// ---------------- LDS-resident per-batch state (CDNA5: 320KB/WGP) ----------
struct WGState {
    float dU[R * R];       // 64 KB fast weights, resident across all 128 steps
    float hb[HB_ELEMS];    // TDM-staged h   (32x512, padded)
    float ab[AB_ELEMS];    // TDM-staged h_att (32x128, padded)
    float ho[R];           // h_att (old)
    float hn[R];           // h_att (new)
    float tre[R];          // trace_e
    float a[R];            // att_in + rec
    float red[8];          // block reduction scratch
    unsigned thkey;        // kwta threshold (monotonic float key)
};

__device__ inline float sigf(float x) { return 1.0f / (1.0f + __expf(-x)); }

__device__ inline unsigned f2key(float f) {
    unsigned u = __float_as_uint(f);
    return (u & 0x80000000u) ? ~u : (u | 0x80000000u);
}
__device__ inline float key2f(unsigned k) {
    unsigned u = (k & 0x80000000u) ? (k ^ 0x80000000u) : ~k;
    return __uint_as_float(u);
}

// Flat shared address: low 32 bits are the LDS byte offset (aperture in high bits)
#define LDS_OFF(p) ((unsigned)(unsigned long long)(p))

// ---------------- Tensor Data Mover: 2D tile (rows x dim0 f32) -> LDS ------
// D# group0: count=1, lds_addr, 57b global_addr, type=2
// D# group1: data_size=4B, pad_enable, pad_interval=256 DW, pad_amount=2 DW,
//            tensor_dim0=dim0, tensor_dim1=rows, tile=whole tensor, stride0=dim0
__device__ inline void tdm_load_2d(const float* gsrc, unsigned lds_byte_off,
                                   unsigned dim0, unsigned rows) {
    unsigned long long ga = (unsigned long long)(size_t)gsrc;
    u32x4 g0;
    g0[0] = 1u;                                         // count=1
    g0[1] = lds_byte_off;                               // lds_addr
    g0[2] = (unsigned)(ga & 0xFFFFFFFFu);               // global_addr[31:0]
    g0[3] = (unsigned)((ga >> 32) & 0x01FFFFFFu) | (2u << 30);  // [56:32] | type=2
    i32x8 g1;
    g1[0] = (int)((2u << 16) | (1u << 20) | (7u << 22) | (1u << 25));
    g1[1] = (int)((dim0 & 0xFFFFu) << 16);              // tensor_dim0 lo16
    g1[2] = (int)((dim0 >> 16) | ((rows & 0xFFFFu) << 16));   // dim0 hi | dim1 lo
    g1[3] = (int)((rows >> 16) | ((dim0 & 0xFFFFu) << 16));   // dim1 hi | tile_dim0
    g1[4] = (int)(rows & 0xFFFFu);                      // tile_dim1, tile_dim2=0
    g1[5] = (int)dim0;                                  // tensor_dim0_stride lo32
    g1[6] = 0;
    g1[7] = 0;
    i32x4 z4 = {0, 0, 0, 0};
#if __clang_major__ >= 23
    i32x8 z8 = {0, 0, 0, 0, 0, 0, 0, 0};
    __builtin_amdgcn_tensor_load_to_lds(g0, g1, z4, z4, z8, 0);
#else
    __builtin_amdgcn_tensor_load_to_lds(g0, g1, z4, z4, 0);
#endif
}

// ---------------- device-wide barrier (32 resident WGs) --------------------
__device__ inline void grid_sync(unsigned* cnt, unsigned* gen) {
    __syncthreads();
    if (threadIdx.x == 0) {
        __threadfence();
        unsigned g = __hip_atomic_load(gen, __ATOMIC_RELAXED, __HIP_MEMORY_SCOPE_AGENT);
        unsigned arrived =
            __hip_atomic_fetch_add(cnt, 1u, __ATOMIC_ACQ_REL, __HIP_MEMORY_SCOPE_AGENT) + 1u;
        if (arrived == NWG) {
            __hip_atomic_store(cnt, 0u, __ATOMIC_RELAXED, __HIP_MEMORY_SCOPE_AGENT);
            __hip_atomic_fetch_add(gen, 1u, __ATOMIC_RELEASE, __HIP_MEMORY_SCOPE_AGENT);
        } else {
            while (__hip_atomic_load(gen, __ATOMIC_ACQUIRE, __HIP_MEMORY_SCOPE_AGENT) == g) {
                __builtin_amdgcn_s_sleep(2);
            }
        }
        __threadfence();
    }
    __syncthreads();
}

// ---------------- block-wide sum (wave32 shuffle + LDS) --------------------
__device__ inline float block_sum(float v, float* scratch) {
    #pragma unroll
    for (int o = 16; o > 0; o >>= 1) v += __shfl_down(v, o, 32);
    int wid = threadIdx.x >> 5, lane = threadIdx.x & 31;
    __syncthreads();
    if (lane == 0) scratch[wid] = v;
    __syncthreads();
    float r = scratch[0];
    #pragma unroll
    for (int w = 1; w < NTHREADS / 32; ++w) r += scratch[w];
    return r;
}

// ---------------- f32 WMMA 16x16x4, 2 N-tiles per wave-job -----------------
// A-frag: lanes 0-15 (M=lane, K=k..k+1), lanes 16-31 (K=k+2..k+3).
// B-frag: lanes 0-15 (N=lane, K=k..k+1), lanes 16-31 (K=k+2..k+3). B = W^T.
template <int KD>
__device__ inline void wmma_pair_g(const float* __restrict__ A,
                                   const float* __restrict__ W,
                                   int m0, int n0, v8f& acc0, v8f& acc1) {
    int lane = threadIdx.x & 31;
    int koff = (lane >> 4) << 1;
    const float* ap  = A + (size_t)(m0 + (lane & 15)) * KD + koff;
    const float* bp0 = W + (size_t)(n0 + (lane & 15)) * KD + koff;
    const float* bp1 = bp0 + (size_t)16 * KD;
    for (int k = 0; k < KD; k += 4) {
        v2f av = *reinterpret_cast<const v2f*>(ap + k);
        v2f b0 = *reinterpret_cast<const v2f*>(bp0 + k);
        v2f b1 = *reinterpret_cast<const v2f*>(bp1 + k);
        acc0 = __builtin_amdgcn_wmma_f32_16x16x4_f32(false, av, false, b0,
                                                     (short)0, acc0, false, false);
        acc1 = __builtin_amdgcn_wmma_f32_16x16x4_f32(false, av, false, b1,
                                                     (short)0, acc1, false, false);
    }
}

template <int KD>
__device__ inline void wmma_pair_l(const float* __restrict__ Al,   // padded LDS
                                   const float* __restrict__ W,
                                   int m0, int n0, v8f& acc0, v8f& acc1) {
    int lane = threadIdx.x & 31;
    int koff = (lane >> 4) << 1;
    int pbase = (m0 + (lane & 15)) * KD + koff;
    const float* bp0 = W + (size_t)(n0 + (lane & 15)) * KD + koff;
    const float* bp1 = bp0 + (size_t)16 * KD;
    for (int k = 0; k < KD; k += 4) {
        int p = pbase + k;
        v2f av = *reinterpret_cast<const v2f*>(Al + PADIDX(p));  // ds_load_b64
        v2f b0 = *reinterpret_cast<const v2f*>(bp0 + k);
        v2f b1 = *reinterpret_cast<const v2f*>(bp1 + k);
        acc0 = __builtin_amdgcn_wmma_f32_16x16x4_f32(false, av, false, b0,
                                                     (short)0, acc0, false, false);
        acc1 = __builtin_amdgcn_wmma_f32_16x16x4_f32(false, av, false, b1,
                                                     (short)0, acc1, false, false);
    }
}

// mod GEMM (N=131): clamp B row so no OOB weight loads; stores guarded.
__device__ inline v8f wmma_tile_mod(const float* __restrict__ A,
                                    const float* __restrict__ W,
                                    int m0, int n0) {
    int lane = threadIdx.x & 31;
    int koff = (lane >> 4) << 1;
    int n = n0 + (lane & 15); if (n > MODN - 1) n = MODN - 1;
    const float* ap = A + (size_t)(m0 + (lane & 15)) * H + koff;
    const float* bp = W + (size_t)n * H + koff;
    v8f acc = {};
    for (int k = 0; k < H; k += 4) {
        v2f av = *reinterpret_cast<const v2f*>(ap + k);
        v2f bv = *reinterpret_cast<const v2f*>(bp + k);
        acc = __builtin_amdgcn_wmma_f32_16x16x4_f32(false, av, false, bv,
                                                    (short)0, acc, false, false);
    }
    return acc;
}

// C/D layout: VGPR r holds (M=r, N=lane) lanes 0-15, (M=8+r, N=lane-16) 16-31
__device__ inline void store_tile(float* __restrict__ C, int ldc, int m0, int n0,
                                  v8f acc, const float* __restrict__ bias, int ncols) {
    int lane = threadIdx.x & 31;
    int col = n0 + (lane & 15);
    int rbase = (lane < 16) ? 0 : 8;
    if (col < ncols) {
        float bv = bias[col];
        #pragma unroll
        for (int r = 0; r < 8; ++r)
            C[(size_t)(m0 + rbase + r) * ldc + col] = acc[r] + bv;
    }
}

__device__ inline float ln_apply(float x, float mu, float rs, float g, float b) {
    return g * (x - mu) * rs + b;
}

// ---------------------------------------------------------------------------
__global__ void sgru_sync_init(unsigned* s) {
    if (threadIdx.x < 2) s[threadIdx.x] = 0u;
}

__global__ __launch_bounds__(NTHREADS)
void sgru_persistent(
    const float* __restrict__ x,
    const float* __restrict__ h0,  const float* __restrict__ v0,
    const float* __restrict__ dU0, const float* __restrict__ tre0,
    const float* __restrict__ trE0,const float* __restrict__ hatt0,
    const float* __restrict__ W_x2h,  const float* __restrict__ b_x2h,
    const float* __restrict__ W_h2h,  const float* __restrict__ b_h2h,
    const float* __restrict__ W_att2h,const float* __restrict__ b_att2h,
    const float* __restrict__ g_lnx,  const float* __restrict__ b_lnx,
    const float* __restrict__ g_lnh,  const float* __restrict__ b_lnh,
    const float* __restrict__ g_lnatt,const float* __restrict__ b_lnatt,
    const float* __restrict__ W_h2mod,const float* __restrict__ b_h2mod,
    const float* __restrict__ lambd, const float* __restrict__ eta,
    const float* __restrict__ tau_att,
    float* __restrict__ out, float* __restrict__ ws, unsigned* __restrict__ sync)
{
    __shared__ WGState S;

    const int b     = blockIdx.x;                       // batch owned by this WG
    const int gwave = blockIdx.x * (NTHREADS / 32) + (threadIdx.x >> 5);

    // workspace layout (floats)
    float* wsWx   = ws;                  // 32*1536 raw pre-LN
    float* wsWh   = wsWx   + B * H3;
    float* wsWatt = wsWh   + B * H3;
    float* wsH    = wsWatt + B * H3;     // 32*512
    float* wsV    = wsH    + B * H;
    float* wsHatt = wsV    + B * H;      // 32*128
    float* wsMod  = wsHatt + B * R;      // 32*131

    // output layout (flat, reference return order)
    float* out_outs = out;
    float* out_mods = out_outs + (size_t)T_STEPS * B * H;
    float* out_h    = out_mods + (size_t)T_STEPS * B * MODN;
    float* out_v    = out_h    + B * H;
    float* out_dU   = out_v    + B * H;
    float* out_tre  = out_dU   + (size_t)B * R * R;
    float* out_trE  = out_tre  + B * R;
    float* out_hatt = out_trE  + (size_t)B * R * R;

    // tr_E state lives directly in its output slot (streamed once per step)
    float* trEb = out_trE + (size_t)b * R * R;

    unsigned* cnt = sync;
    unsigned* gen = sync + 1;

    const float lam = sigf(lambd[0]);
    const float et  = sigf(eta[0]);
    const float tau = sigf(tau_att[0]);

    // ---- init: per-batch fast weights into LDS, recurrent state into ws ----
    for (int j = threadIdx.x; j < H; j += NTHREADS) {
        wsH[b * H + j] = h0[b * H + j];
        wsV[b * H + j] = v0[b * H + j];
    }
    for (int j = threadIdx.x; j < R; j += NTHREADS) {
        wsHatt[b * R + j] = hatt0[b * R + j];
        S.tre[j]          = tre0[b * R + j];
    }
    for (int idx = threadIdx.x; idx < R * R; idx += NTHREADS) {
        S.dU[idx]  = dU0 [(size_t)b * R * R + idx];
        trEb[idx]  = trE0[(size_t)b * R * R + idx];
    }
    grid_sync(cnt, gen);

    for (int t = 0; t < T_STEPS; ++t) {
        // ===== TDM: stage A-operands (h, h_att) into LDS, padded ============
        if (threadIdx.x < 32) {         // wave 0 only: TDM ignores EXEC
            tdm_load_2d(wsH,    LDS_OFF(&S.hb[0]), H, B);
            tdm_load_2d(wsHatt, LDS_OFF(&S.ab[0]), R, B);
            __builtin_amdgcn_s_wait_tensorcnt(0);
        }
        __syncthreads();

        // ===== phase 1: three WMMA GEMMs into (32 x 1536), 2 N-tiles/job ====
        for (int job = gwave; job < 3 * 96; job += NWAVES_TOTAL) {
            int mat = job / 96;
            int tt  = job % 96;
            int mt = tt / 48, np = tt % 48;
            int m0 = mt * 16, n0 = np * 32;
            v8f a0 = {}, a1 = {};
            if (mat == 0) {
                wmma_pair_g<IN_DIM>(x + (size_t)t * B * IN_DIM, W_x2h, m0, n0, a0, a1);
                store_tile(wsWx, H3, m0, n0,      a0, b_x2h, H3);
                store_tile(wsWx, H3, m0, n0 + 16, a1, b_x2h, H3);
            } else if (mat == 1) {
                wmma_pair_l<H>(S.hb, W_h2h, m0, n0, a0, a1);
                store_tile(wsWh, H3, m0, n0,      a0, b_h2h, H3);
                store_tile(wsWh, H3, m0, n0 + 16, a1, b_h2h, H3);
            } else {
                wmma_pair_l<R>(S.ab, W_att2h, m0, n0, a0, a1);
                store_tile(wsWatt, H3, m0, n0,      a0, b_att2h, H3);
                store_tile(wsWatt, H3, m0, n0 + 16, a1, b_att2h, H3);
            }
        }
        grid_sync(cnt, gen);

        // ===== phase 1b: per-batch LN stats (rows of 1536) + gate fusion =====
        const float* rows[3] = { wsWx + (size_t)b * H3, wsWh + (size_t)b * H3,
                                 wsWatt + (size_t)b * H3 };
        float mu[3], rs[3];
        for (int mtx = 0; mtx < 3; ++mtx) {
            float s = 0.f, s2 = 0.f;
            const float* row = rows[mtx];
            for (int j = threadIdx.x; j < H3; j += NTHREADS) {
                float vv = row[j]; s += vv; s2 += vv * vv;
            }
            s  = block_sum(s,  S.red);
            s2 = block_sum(s2, S.red);
            float m = s * (1.0f / H3);
            float var = s2 * (1.0f / H3) - m * m;
            mu[mtx] = m; rs[mtx] = rsqrtf(var + 1e-5f);
        }
        for (int j = threadIdx.x; j < H; j += NTHREADS) {
            float zx  = ln_apply(rows[0][j],         mu[0], rs[0], g_lnx[j],         b_lnx[j]);
            float rx  = ln_apply(rows[0][j + H],     mu[0], rs[0], g_lnx[j + H],     b_lnx[j + H]);
            float dvx = ln_apply(rows[0][j + 2 * H], mu[0], rs[0], g_lnx[j + 2 * H], b_lnx[j + 2 * H]);
            float zh  = ln_apply(rows[1][j],         mu[1], rs[1], g_lnh[j],         b_lnh[j]);
            float rh  = ln_apply(rows[1][j + H],     mu[1], rs[1], g_lnh[j + H],     b_lnh[j + H]);
            float dvh = ln_apply(rows[1][j + 2 * H], mu[1], rs[1], g_lnh[j + 2 * H], b_lnh[j + 2 * H]);
            float za  = ln_apply(rows[2][j],         mu[2], rs[2], g_lnatt[j],         b_lnatt[j]);
            float ra  = ln_apply(rows[2][j + H],     mu[2], rs[2], g_lnatt[j + H],     b_lnatt[j + H]);
            float dva = ln_apply(rows[2][j + 2 * H], mu[2], rs[2], g_lnatt[j + 2 * H], b_lnatt[j + 2 * H]);
            float z  = sigf(zx + zh + za);
            float r_ = sigf(rx + rh + ra);
            float dv = dvx + dvh + dva;
            float vn = (1.f - z) * wsV[b * H + j] + z * r_ * dv;
            wsV[b * H + j] = vn;
            float hn = fmaxf(vn, 0.f);
            wsH[b * H + j] = hn;
            out_outs[(size_t)t * B * H + (size_t)b * H + j] = hn;
        }
        grid_sync(cnt, gen);

        // ===== phase 2: mod = relu(h @ W_h2mod^T + b), WMMA 2x9 tiles ========
        for (int job = gwave; job < 2 * 9; job += NWAVES_TOTAL) {
            int mt = job / 9, nt = job % 9;
            int m0 = mt * 16, n0 = nt * 16;
            v8f acc = wmma_tile_mod(wsH, W_h2mod, m0, n0);
            int lane = threadIdx.x & 31;
            int col = n0 + (lane & 15);
            int rbase = (lane < 16) ? 0 : 8;
            if (col < MODN) {
                float bv = b_h2mod[col];
                #pragma unroll
                for (int r = 0; r < 8; ++r)
                    wsMod[(size_t)(m0 + rbase + r) * MODN + col] = fmaxf(acc[r] + bv, 0.f);
            }
        }
        grid_sync(cnt, gen);

        // ===== phase 3: per-batch attention / kwta / fast-weight update ======
        const float* modb = wsMod + (size_t)b * MODN;
        for (int j = threadIdx.x; j < MODN; j += NTHREADS)
            out_mods[(size_t)t * B * MODN + (size_t)b * MODN + j] = modb[j];
        float d_ = sigf(modb[0]);
        float s_ = sigf(modb[1]);
        float m_ = tanhf(modb[2]);
        if (threadIdx.x < R) S.ho[threadIdx.x] = wsHatt[b * R + threadIdx.x];
        if (threadIdx.x == 0) S.thkey = 0xFFFFFFFFu;
        __syncthreads();
        if (threadIdx.x < R) {                       // rec = dU @ h_att (LDS)
            float accr = 0.f;
            const float* rowdU = S.dU + threadIdx.x * R;
            for (int j = 0; j < R; ++j) accr = fmaf(rowdU[j], S.ho[j], accr);
            S.a[threadIdx.x] = modb[3 + threadIdx.x] + accr;
        }
        __syncthreads();
        if (threadIdx.x < R) {                       // kwta threshold (12th largest)
            float ai = S.a[threadIdx.x];
            int c = 0;
            for (int j = 0; j < R; ++j) c += (S.a[j] > ai) ? 1 : 0;
            if (c <= KTOP - 1) atomicMin(&S.thkey, f2key(ai));
        }
        __syncthreads();
        float th = key2f(S.thkey);
        if (threadIdx.x < R) {
            float ai = S.a[threadIdx.x];
            float kw = (ai >= th) ? ai : 0.f;
            float hn = (1.f - tau) * S.ho[threadIdx.x] + tau * kw;
            S.hn[threadIdx.x] = hn;
            float te = (1.f - d_) * S.tre[threadIdx.x] + d_ * S.ho[threadIdx.x];
            S.tre[threadIdx.x] = te;
            wsHatt[b * R + threadIdx.x] = hn;
        }
        __syncthreads();
        float em = et * m_, kl = 1.f - lam, sm = s_;
        for (int idx = threadIdx.x; idx < R * R; idx += NTHREADS) {
            int i = idx >> 7, j = idx & (R - 1);
            float outer = S.hn[i] * S.tre[j] - S.hn[j] * S.tre[i];
            float tE = (1.f - sm) * trEb[idx] + sm * outer;   // streamed via L2
            trEb[idx] = tE;
            S.dU[idx]  = kl * S.dU[idx] + em * tE;
        }
        grid_sync(cnt, gen);
    }

    // ===== write final states (tr_E already lives in d_out) ================
    for (int j = threadIdx.x; j < H; j += NTHREADS) {
        out_h[b * H + j] = wsH[b * H + j];
        out_v[b * H + j] = wsV[b * H + j];
    }
    for (int j = threadIdx.x; j < R; j += NTHREADS) {
        out_tre[b * R + j]  = S.tre[j];
        out_hatt[b * R + j] = wsHatt[b * R + j];
    }
    for (int idx = threadIdx.x; idx < R * R; idx += NTHREADS)
        out_dU[(size_t)b * R * R + idx] = S.dU[idx];
}

extern "C" void kernel_launch(void* const* d_in, const int* in_sizes, int n_in,
                              void* d_out, int out_size, void* d_ws, size_t ws_size,
                              hipStream_t stream) {
    (void)in_sizes; (void)n_in; (void)out_size; (void)ws_size;
    const float* x       = (const float*)d_in[0];
    const float* h0      = (const float*)d_in[1];
    const float* v0      = (const float*)d_in[2];
    const float* dU0     = (const float*)d_in[3];
    const float* tre0    = (const float*)d_in[4];
    const float* trE0    = (const float*)d_in[5];
    const float* hatt0   = (const float*)d_in[6];
    const float* W_x2h   = (const float*)d_in[7];
    const float* b_x2h   = (const float*)d_in[8];
    const float* W_h2h   = (const float*)d_in[9];
    const float* b_h2h   = (const float*)d_in[10];
    const float* W_att2h = (const float*)d_in[11];
    const float* b_att2h = (const float*)d_in[12];
    const float* g_lnx   = (const float*)d_in[13];
    const float* b_lnx   = (const float*)d_in[14];
    const float* g_lnh   = (const float*)d_in[15];
    const float* b_lnh   = (const float*)d_in[16];
    const float* g_lnatt = (const float*)d_in[17];
    const float* b_lnatt = (const float*)d_in[18];
    const float* W_h2mod = (const float*)d_in[19];
    const float* b_h2mod = (const float*)d_in[20];
    const float* lambd   = (const float*)d_in[21];
    const float* eta     = (const float*)d_in[22];
    const float* tau     = (const float*)d_in[23];

    float* ws = (float*)d_ws;
    size_t ws_floats = (size_t)3 * B * H3 + 2 * B * H + B * R + B * MODN;
    unsigned* sync = (unsigned*)(ws + ws_floats);

    sgru_sync_init<<<1, 32, 0, stream>>>(sync);
    sgru_persistent<<<dim3(NWG), dim3(NTHREADS), 0, stream>>>(
        x, h0, v0, dU0, tre0, trE0, hatt0,
        W_x2h, b_x2h, W_h2h, b_h2h, W_att2h, b_att2h,
        g_lnx, b_lnx, g_lnh, b_lnh, g_lnatt, b_lnatt,
        W_h2mod, b_h2mod, lambd, eta, tau,
        (float*)d_out, ws, sync);
}